// reblurWithKernel_25589415149664
// MI455X (gfx1250) — compile-verified
//
#include <hip/hip_runtime.h>

// Per-pixel 3x3 convolution, edge-replicate padding.
// inputImg: [1,3,1024,1024] f32, Kernels: [9,1024,1024] f32 -> out [1,3,1024,1024] f32.
// Memory-bound (~60 MB @ 23.3 TB/s => ~2.6 us floor). Strategy: async global->LDS
// staging of stencil rows (CDNA5 ASYNCcnt path), b128 coalesced accesses,
// non-temporal hints on the single-use Kernels tensor and the output.
// One block = one full image row (1024 px), 256 threads (8 waves), 4 px/thread.

#define NTHREADS 256
#define ROWSTRIDE 1032  // floats per staged LDS row; interior at +4 => 16B aligned

typedef __attribute__((ext_vector_type(4))) float floatx4;
typedef __attribute__((ext_vector_type(4))) int   intx4;

#define GAS __attribute__((address_space(1)))
#define LAS __attribute__((address_space(3)))

static __device__ __forceinline__ void async_copy_b128(void* lds, const void* gsrc) {
#if __has_builtin(__builtin_amdgcn_global_load_async_to_lds_b128)
  __builtin_amdgcn_global_load_async_to_lds_b128(
      (GAS intx4*)gsrc, (LAS intx4*)lds, 0, 0);
#else
  asm volatile("global_load_async_to_lds_b128 %0, %1, off"
               :
               : "v"((LAS void*)lds), "v"(gsrc)
               : "memory");
#endif
}

static __device__ __forceinline__ void wait_async_zero() {
#if __has_builtin(__builtin_amdgcn_s_wait_asynccnt)
  __builtin_amdgcn_s_wait_asynccnt(0);
#else
  asm volatile("s_wait_asynccnt 0" ::: "memory");
#endif
}

__global__ __launch_bounds__(NTHREADS) void reblur_perpixel3x3_kernel(
    const float* __restrict__ x,     // [3,H,W]
    const float* __restrict__ kern,  // [9,H,W]
    float* __restrict__ out) {       // [3,H,W]
  constexpr int H = 1024;
  constexpr int W = 1024;
  constexpr int HW = H * W;

  const int t = threadIdx.x;         // 0..255, each owns 4 consecutive pixels
  const int h = blockIdx.x;          // one block per image row

  __shared__ float s_x[9 * ROWSTRIDE];  // 9 rows (3ch x 3 stencil rows), padded

  // ---- Stage interior of the 9 rows via async copy engine (ASYNCcnt) ----
  // rc = c*3 + r ; LDS interior float index 4..4+W-1 ; global cols 0..W-1.
#pragma unroll
  for (int rc = 0; rc < 9; ++rc) {
    const int c = rc / 3;
    const int r = rc % 3;
    int hr = h + r - 1;
    hr = hr < 0 ? 0 : (hr >= H ? H - 1 : hr);   // edge-replicate rows
    const float* g = x + c * HW + hr * W + 4 * t;
    async_copy_b128(&s_x[rc * ROWSTRIDE + 4 + 4 * t], g);
  }

  // ---- Halo columns = true image edges (replicate col 0 and col W-1) ----
  if (t < 9) {
    const int rc = t, c = rc / 3, r = rc % 3;
    int hr = h + r - 1;
    hr = hr < 0 ? 0 : (hr >= H ? H - 1 : hr);
    s_x[rc * ROWSTRIDE + 3] = x[c * HW + hr * W + 0];
  } else if (t >= 16 && t < 25) {
    const int rc = t - 16, c = rc / 3, r = rc % 3;
    int hr = h + r - 1;
    hr = hr < 0 ? 0 : (hr >= H ? H - 1 : hr);
    s_x[rc * ROWSTRIDE + 4 + W] = x[c * HW + hr * W + (W - 1)];
  }

  // ---- Per-pixel kernels: 9 taps x 4 pixels, streamed (used exactly once) ----
  floatx4 kv[9];
  const int pb = h * W + 4 * t;
#pragma unroll
  for (int q = 0; q < 9; ++q) {
    kv[q] = __builtin_nontemporal_load((const floatx4*)(kern + q * HW + pb));
  }

  wait_async_zero();   // our wave's async LDS writes done
  __syncthreads();     // all waves' staging visible

  // ---- 9-tap weighted sum for 3 channels x 4 pixels ----
#pragma unroll
  for (int c = 0; c < 3; ++c) {
    floatx4 acc = {0.0f, 0.0f, 0.0f, 0.0f};
#pragma unroll
    for (int r = 0; r < 3; ++r) {
      const float* row = &s_x[(c * 3 + r) * ROWSTRIDE + 3 + 4 * t];
      float xv[6];
#pragma unroll
      for (int u = 0; u < 6; ++u) xv[u] = row[u];
#pragma unroll
      for (int j = 0; j < 3; ++j) {
        const floatx4 kq = kv[3 * r + j];
#pragma unroll
        for (int i = 0; i < 4; ++i)
          acc[i] = fmaf(kq[i], xv[i + j], acc[i]);
      }
    }
    __builtin_nontemporal_store(acc, (floatx4*)(out + c * HW + pb));
  }
}

extern "C" void kernel_launch(void* const* d_in, const int* in_sizes, int n_in,
                              void* d_out, int out_size, void* d_ws, size_t ws_size,
                              hipStream_t stream) {
  (void)in_sizes; (void)n_in; (void)out_size; (void)d_ws; (void)ws_size;
  const float* x    = (const float*)d_in[0];  // inputImg [1,3,1024,1024]
  const float* kern = (const float*)d_in[1];  // Kernels  [9,1024,1024]
  float* out        = (float*)d_out;          // [1,3,1024,1024]

  dim3 grid(1024, 1, 1);       // one block per image row
  dim3 block(NTHREADS, 1, 1);
  reblur_perpixel3x3_kernel<<<grid, block, 0, stream>>>(x, kern, out);
}